// MultiResolutionHashEncoder_1400159339018
// MI455X (gfx1250) — compile-verified
//
#include <hip/hip_runtime.h>
#include <math.h>

#define NLV 12
#define FPL 8
#define PRIME1 2654435761u
#define STAGED 3
#define BLOCK 512
// staged levels 0..2: hash sizes 256, 441, 900 entries; 8 floats each = 2 float4 each
#define SMEM_F4 ((256 + 441 + 900) * 2)   // 3194 vec4 = 51,104 bytes LDS

typedef float v4f __attribute__((ext_vector_type(4)));

struct HashEncParams {
    const float* coords;            // [N,2]
    const float* tables[NLV];       // each [hsize[l], 8]
    float*       out;               // [N,96] encoded  ++  12 x [N,8] level blocks
    int          res[NLV];
    int          hsize[NLV];
    double       hinv[NLV];         // 1.0 / hsize[l]
    int          npts;
};

// Exact 64-bit mod for code < 2^42 via double reciprocal + one-step fixup.
static __device__ __forceinline__ int hmod(unsigned long long code, int h, double hinv) {
    unsigned long long q = (unsigned long long)((double)code * hinv);
    long long r = (long long)(code - q * (unsigned long long)h);
    r = (r < 0) ? (r + h) : r;
    r = (r >= h) ? (r - h) : r;
    return (int)r;
}

static __device__ __forceinline__ void corner_acc(const v4f* tb, int idx, float w,
                                                  v4f& lo, v4f& hi) {
    v4f e0 = tb[2 * idx];
    v4f e1 = tb[2 * idx + 1];
    lo += e0 * w;   // contracts to v_fma under -O3
    hi += e1 * w;
}

__global__ __launch_bounds__(BLOCK) void hashenc_kernel(HashEncParams p) {
    __shared__ v4f smem[SMEM_F4];
    const int tid = threadIdx.x;

    // ---- Stage level 0..2 tables into LDS via CDNA5 async global->LDS DMA ----
    const int f4cnt[STAGED]  = {512, 882, 1800};
    const int f4base[STAGED] = {0, 512, 1394};
#pragma unroll
    for (int l = 0; l < STAGED; ++l) {
        unsigned long long gb = (unsigned long long)p.tables[l];
        unsigned lbase = (unsigned)(unsigned long long)(&smem[f4base[l]]);
        for (int i = tid; i < f4cnt[l]; i += BLOCK) {
            unsigned lds_addr = lbase + (unsigned)i * 16u;   // LDS byte address (VDST)
            unsigned goff     = (unsigned)i * 16u;           // per-lane byte offset (VADDR)
            asm volatile("global_load_async_to_lds_b128 %0, %1, %2"
                         :
                         : "v"(lds_addr), "v"(goff), "s"(gb)
                         : "memory");
        }
    }
    asm volatile("s_wait_asynccnt 0" ::: "memory");
    __syncthreads();

    const int pt = blockIdx.x * BLOCK + tid;
    if (pt >= p.npts) return;

    float2 c = ((const float2*)p.coords)[pt];
    float cx = fminf(fmaxf(c.x, 0.0f), 1.0f);
    float cy = fminf(fmaxf(c.y, 0.0f), 1.0f);

    const size_t N = (size_t)p.npts;
    float* oenc = p.out + (size_t)pt * (NLV * FPL);
    float* olvl = p.out + N * (NLV * FPL) + (size_t)pt * FPL;

#pragma unroll
    for (int l = 0; l < NLV; ++l) {
        const int    rs   = p.res[l];
        const int    h    = p.hsize[l];
        const double hinv = p.hinv[l];

        const float rf = (float)rs;
        float sx = cx * rf, sy = cy * rf;
        float bxf = floorf(sx), byf = floorf(sy);
        float fx = sx - bxf, fy = sy - byf;
        int bx = (int)bxf, by = (int)byf;
        int rm1 = rs - 1;
        int x0 = min(max(bx, 0), rm1);
        int x1 = min(bx + 1, rm1);
        int y0 = min(max(by, 0), rm1);
        int y1 = min(by + 1, rm1);

        unsigned long long hy0 = (unsigned long long)(unsigned)y0 * PRIME1;
        unsigned long long hy1 = (unsigned long long)(unsigned)y1 * PRIME1;
        int i00 = hmod(hy0 ^ (unsigned)x0, h, hinv);
        int i01 = hmod(hy1 ^ (unsigned)x0, h, hinv);
        int i10 = hmod(hy0 ^ (unsigned)x1, h, hinv);
        int i11 = hmod(hy1 ^ (unsigned)x1, h, hinv);

        float w00 = (1.0f - fx) * (1.0f - fy);
        float w01 = (1.0f - fx) * fy;
        float w10 = fx * (1.0f - fy);
        float w11 = fx * fy;

        v4f lo = (v4f)(0.0f);
        v4f hi = (v4f)(0.0f);

        if (l < STAGED) {
            const int sb[STAGED] = {0, 512, 1394};
            const v4f* tb = &smem[sb[l]];
            corner_acc(tb, i00, w00, lo, hi);
            corner_acc(tb, i01, w01, lo, hi);
            corner_acc(tb, i10, w10, lo, hi);
            corner_acc(tb, i11, w11, lo, hi);
        } else {
            const v4f* __restrict__ tb = (const v4f*)p.tables[l];
            corner_acc(tb, i00, w00, lo, hi);
            corner_acc(tb, i01, w01, lo, hi);
            corner_acc(tb, i10, w10, lo, hi);
            corner_acc(tb, i11, w11, lo, hi);
        }

        // Streaming (non-temporal) stores: output is 402MB, > 2x L2 — keep L2 for tables.
        v4f* oe = (v4f*)(oenc + l * FPL);
        __builtin_nontemporal_store(lo, oe);
        __builtin_nontemporal_store(hi, oe + 1);

        v4f* ol = (v4f*)(olvl + (size_t)l * N * FPL);
        __builtin_nontemporal_store(lo, ol);
        __builtin_nontemporal_store(hi, ol + 1);
    }
}

extern "C" void kernel_launch(void* const* d_in, const int* in_sizes, int n_in,
                              void* d_out, int out_size, void* d_ws, size_t ws_size,
                              hipStream_t stream) {
    (void)n_in; (void)out_size; (void)d_ws; (void)ws_size;

    HashEncParams prm;
    prm.coords = (const float*)d_in[0];
    for (int i = 0; i < NLV; ++i) prm.tables[i] = (const float*)d_in[1 + i];
    prm.out  = (float*)d_out;
    prm.npts = in_sizes[0] / 2;

    // Mirror the Python level-parameter computation in double precision.
    double s = exp((log(512.0) - log(16.0)) / 11.0);
    for (int l = 0; l < NLV; ++l) {
        int r = (int)(16.0 * pow(s, (double)l));   // truncation, same as Python int()
        if (r < 2) r = 2;
        long long hh = (long long)r * (long long)r;
        if (hh > 131072) hh = 131072;
        prm.res[l]   = r;
        prm.hsize[l] = (int)hh;
        prm.hinv[l]  = 1.0 / (double)hh;
    }

    int nblocks = (prm.npts + BLOCK - 1) / BLOCK;
    hipLaunchKernelGGL(hashenc_kernel, dim3(nblocks), dim3(BLOCK), 0, stream, prm);
}